// MultiBoxLoss_76828374991091
// MI455X (gfx1250) — compile-verified
//
#include <hip/hip_runtime.h>
#include <hip/hip_bf16.h>
#include <stdint.h>

#define BB 32
#define PP 8732
#define CC 81
#define NPRIOR (BB * PP)          // 279424
#define WPB 8                     // waves per block in k1
#define NB1 (NPRIOR / WPB)        // 34928 blocks, exact

// ---------- wave32 reductions ----------
__device__ __forceinline__ float waveMax32(float v) {
#pragma unroll
    for (int off = 16; off >= 1; off >>= 1)
        v = fmaxf(v, __shfl_xor(v, off, 32));
    return v;
}
__device__ __forceinline__ float waveSum32(float v) {
#pragma unroll
    for (int off = 16; off >= 1; off >>= 1)
        v += __shfl_xor(v, off, 32);
    return v;
}

// ============================================================
// Kernel 1: per-prior logsumexp / nll / smooth-L1 partials.
// One wave (32 lanes) per prior; 8 waves per block; exact grid.
// ============================================================
__global__ __launch_bounds__(256) void mbox_k1_perprior(
        const float* __restrict__ conf,   // [B,P,C]
        const float* __restrict__ locd,   // [B,P,4]
        const int*   __restrict__ tgt,    // [B,P]
        const float* __restrict__ loct,   // [B,P,4]
        float* __restrict__ rank,         // [B,P] out: pos?0:nll
        float* __restrict__ partPos,      // [NB1] out
        float* __restrict__ partLoc)      // [NB1] out
{
    const int wave = threadIdx.x >> 5;
    const int lane = threadIdx.x & 31;
    const int i    = blockIdx.x * WPB + wave;          // prior index (exact fit)

    const float* base = conf + (size_t)i * CC;
    // 81 = 32 + 32 + 17 coalesced lane-strided loads
    const float x0 = base[lane];
    const float x1 = base[lane + 32];
    const float x2 = (lane < 17) ? base[lane + 64] : -3.402823466e38f;

    const float M = waveMax32(fmaxf(x0, fmaxf(x1, x2)));
    float s = __expf(x0 - M) + __expf(x1 - M) +
              ((lane < 17) ? __expf(x2 - M) : 0.0f);
    s = waveSum32(s);

    const int  t   = tgt[i];                 // broadcast load (same addr all lanes)
    const bool pos = (t > 0);
    const float nll = M + __logf(s) - base[t];   // lse - target logit

    if (lane == 0) rank[i] = pos ? 0.0f : nll;

    float posC = (lane == 0 && pos) ? nll : 0.0f;

    // smooth L1 on lanes 0..3 (one coordinate each), only if positive
    float locC = 0.0f;
    if (lane < 4 && pos) {
        const float d  = locd[(size_t)i * 4 + lane] - loct[(size_t)i * 4 + lane];
        const float ad = fabsf(d);
        locC = (ad < 1.0f) ? 0.5f * d * d : (ad - 0.5f);
    }
    locC += __shfl_xor(locC, 1, 32);
    locC += __shfl_xor(locC, 2, 32);         // lane0 now holds wave loc partial

    __shared__ float sp[WPB], sl[WPB];
    if (lane == 0) { sp[wave] = posC; sl[wave] = locC; }
    __syncthreads();
    if (threadIdx.x == 0) {
        float a = 0.0f, l = 0.0f;
#pragma unroll
        for (int w = 0; w < WPB; ++w) { a += sp[w]; l += sl[w]; }
        partPos[blockIdx.x] = a;             // deterministic partials, no atomics
        partLoc[blockIdx.x] = l;
    }
}

// ============================================================
// Kernel 2: per-batch hard-negative mining = sum of top-m of
// loss_rank. Stages the row into LDS via CDNA5 async-to-LDS,
// then 4-pass radix-select on float bit patterns (all >= 0).
// One block of 512 threads per batch.
// ============================================================
__global__ __launch_bounds__(512) void mbox_k2_select(
        const int*   __restrict__ tgt,     // [B,P]
        const float* __restrict__ rank,    // [B,P]
        float* __restrict__ negSum,        // [B] out
        int*   __restrict__ numPos)        // [B] out
{
    __shared__ __align__(16) float vals[PP];   // 34928 B
    __shared__ unsigned cnt[256];
    __shared__ unsigned redu[512];
    __shared__ float    redf[512];
    __shared__ unsigned bc[2];

    const int b   = blockIdx.x;
    const int tid = threadIdx.x;
    const int nt  = blockDim.x;                // 512

    // ---- count positives (num_pos) ----
    unsigned myPos = 0;
    for (int i = tid; i < PP; i += nt) myPos += (tgt[b * PP + i] > 0) ? 1u : 0u;
    redu[tid] = myPos; __syncthreads();
    for (int s = 256; s > 0; s >>= 1) {
        if (tid < s) redu[tid] += redu[tid + s];
        __syncthreads();
    }
    const int np = (int)redu[0];
    if (tid == 0) numPos[b] = np;

    int m = 3 * np;
    if (m > PP - 1)  m = PP - 1;
    if (m > PP - np) m = PP - np;              // only negatives contribute

    // ---- async-stage rank row -> LDS (GLOBAL_LOAD_ASYNC_TO_LDS_B128) ----
    const float* row = rank + (size_t)b * PP;  // 16B-aligned (P*4 % 16 == 0)
    for (int q = tid; q < PP / 4; q += nt) {   // 2183 exact float4s
        unsigned lds_off = (unsigned)(uintptr_t)(&vals[q * 4]);
        const float* gp  = row + q * 4;
        asm volatile("global_load_async_to_lds_b128 %0, %1, off"
                     :: "v"(lds_off), "v"(gp) : "memory");
    }
    asm volatile("s_wait_asynccnt 0" ::: "memory");
    __syncthreads();

    if (m < 1) {                               // uniform over block
        if (tid == 0) negSum[b] = 0.0f;
        return;
    }

    // ---- 4x8-bit radix select for the m-th largest value T ----
    unsigned prefix = 0, remaining = (unsigned)m;
    for (int pass = 0; pass < 4; ++pass) {
        const int shift = 24 - 8 * pass;
        if (tid < 256) cnt[tid] = 0;
        __syncthreads();
        for (int i = tid; i < PP; i += nt) {
            const unsigned u = __float_as_uint(vals[i]);
            const bool inClass =
                (pass == 0) || ((u >> (shift + 8)) == (prefix >> (shift + 8)));
            if (inClass) atomicAdd(&cnt[(u >> shift) & 255u], 1u);
        }
        __syncthreads();
        if (tid == 0) {
            unsigned acc = 0; int bsel = 0;
            for (int j = 255; j >= 0; --j) {
                const unsigned c = cnt[j];
                if (acc + c >= remaining) { bsel = j; break; }
                acc += c;
            }
            bc[0] = (unsigned)bsel;
            bc[1] = remaining - acc;
        }
        __syncthreads();
        prefix |= bc[0] << shift;
        remaining = bc[1];
        __syncthreads();
    }

    // ---- sum of top-m = sum(v > T) + (m - count(v > T)) * T ----
    const unsigned T  = prefix;
    const float    Tf = __uint_as_float(T);
    unsigned gcnt = 0; float gsum = 0.0f;
    for (int i = tid; i < PP; i += nt) {
        const float v = vals[i];
        if (__float_as_uint(v) > T) { ++gcnt; gsum += v; }
    }
    redu[tid] = gcnt; redf[tid] = gsum; __syncthreads();
    for (int s = 256; s > 0; s >>= 1) {
        if (tid < s) { redu[tid] += redu[tid + s]; redf[tid] += redf[tid + s]; }
        __syncthreads();
    }
    if (tid == 0)
        negSum[b] = redf[0] + (float)(m - (int)redu[0]) * Tf;
}

// ============================================================
// Kernel 3: finalize — fold partials, divide by N = sum(num_pos)
// ============================================================
__global__ __launch_bounds__(512) void mbox_k3_finalize(
        const float* __restrict__ partPos, const float* __restrict__ partLoc,
        const float* __restrict__ negSum,  const int* __restrict__ numPos,
        float* __restrict__ out)
{
    __shared__ float rp[512], rl[512];
    const int tid = threadIdx.x;
    float a = 0.0f, l = 0.0f;
    for (int i = tid; i < NB1; i += 512) { a += partPos[i]; l += partLoc[i]; }
    rp[tid] = a; rl[tid] = l; __syncthreads();
    for (int s = 256; s > 0; s >>= 1) {
        if (tid < s) { rp[tid] += rp[tid + s]; rl[tid] += rl[tid + s]; }
        __syncthreads();
    }
    if (tid == 0) {
        float neg = 0.0f; long long N = 0;
        for (int b = 0; b < BB; ++b) { neg += negSum[b]; N += numPos[b]; }
        const float Nf = (float)N;
        out[0] = (rp[0] + neg) / Nf;   // loss_c / N
        out[1] = rl[0] / Nf;           // loss_l / N
    }
}

extern "C" void kernel_launch(void* const* d_in, const int* in_sizes, int n_in,
                              void* d_out, int out_size, void* d_ws, size_t ws_size,
                              hipStream_t stream) {
    const float* conf = (const float*)d_in[0];   // [32,8732,81] f32
    const float* locd = (const float*)d_in[1];   // [32,8732,4]  f32
    const int*   tgt  = (const int*)d_in[2];     // [32,8732]    int
    const float* loct = (const float*)d_in[3];   // [32,8732,4]  f32
    float* out = (float*)d_out;                  // 2 scalars

    char* ws = (char*)d_ws;
    float* rank    = (float*)ws;  ws += sizeof(float) * NPRIOR;   // 1.12 MB
    float* partPos = (float*)ws;  ws += sizeof(float) * NB1;
    float* partLoc = (float*)ws;  ws += sizeof(float) * NB1;
    float* negSum  = (float*)ws;  ws += sizeof(float) * BB;
    int*   numPos  = (int*)ws;

    hipLaunchKernelGGL(mbox_k1_perprior, dim3(NB1), dim3(256), 0, stream,
                       conf, locd, tgt, loct, rank, partPos, partLoc);
    hipLaunchKernelGGL(mbox_k2_select, dim3(BB), dim3(512), 0, stream,
                       tgt, rank, negSum, numPos);
    hipLaunchKernelGGL(mbox_k3_finalize, dim3(1), dim3(512), 0, stream,
                       partPos, partLoc, negSum, numPos, out);
}